// RecSysGNN_53077205844499
// MI455X (gfx1250) — compile-verified
//
#include <hip/hip_runtime.h>

#define LATENT 64

typedef __attribute__((ext_vector_type(2))) float v2f;
typedef __attribute__((ext_vector_type(8))) float v8f;

// ---------------------------------------------------------------------------
// Zero a contiguous float region (deg + layer buffers).
// ---------------------------------------------------------------------------
__global__ void zero_kernel(float* __restrict__ p, int n) {
  int i = blockIdx.x * blockDim.x + threadIdx.x;
  if (i < n) p[i] = 0.0f;
}

// ---------------------------------------------------------------------------
// deg[dst[e]] += 1.0f  (no-return f32 atomic -> resolves in L2 atomic units)
// ---------------------------------------------------------------------------
__global__ void degree_kernel(const int* __restrict__ dst, float* __restrict__ deg, int E) {
  int e = blockIdx.x * blockDim.x + threadIdx.x;
  if (e < E) atomicAdd(&deg[dst[e]], 1.0f);
}

// ---------------------------------------------------------------------------
// deg[i] = deg[i] > 0 ? rsqrt(deg[i]) : 0   (in place; counts are >= 1 so the
// reference's max(deg,1) guard is implied)
// ---------------------------------------------------------------------------
__global__ void invsqrt_kernel(float* __restrict__ deg, int n) {
  int i = blockIdx.x * blockDim.x + threadIdx.x;
  if (i < n) {
    float d = deg[i];
    deg[i] = d > 0.0f ? rsqrtf(d) : 0.0f;
  }
}

// ---------------------------------------------------------------------------
// norm[e] = dinv[src[e]] * dinv[dst[e]]
// ---------------------------------------------------------------------------
__global__ void norm_kernel(const int* __restrict__ src, const int* __restrict__ dst,
                            const float* __restrict__ dinv, float* __restrict__ nrm, int E) {
  int e = blockIdx.x * blockDim.x + threadIdx.x;
  if (e < E) nrm[e] = dinv[src[e]] * dinv[dst[e]];
}

// ---------------------------------------------------------------------------
// One propagation layer: eout[dst[e]] += norm[e] * ein[src[e]]   (row of 64 f32)
// 16 threads per edge, float4 per thread: 16 consecutive lanes read one
// 256B-coalesced embedding row; scatter via 4 f32 atomics (L2-resident:
// both embedding buffers fit in the 192MB global L2, so the atomic RMWs and
// gathers never spill to HBM after first touch).
// ---------------------------------------------------------------------------
__global__ void propagate_kernel(const float* __restrict__ ein, float* __restrict__ eout,
                                 const int* __restrict__ src, const int* __restrict__ dst,
                                 const float* __restrict__ nrm, int E) {
  unsigned t = blockIdx.x * blockDim.x + threadIdx.x;
  unsigned e = t >> 4;
  if (e >= (unsigned)E) return;
  unsigned c = (t & 15u) << 2;  // float offset within the 64-float row

  if ((t & 15u) == 0u) __builtin_prefetch(nrm + e + 2048, 0, 0);  // global_prefetch_b8

  int s = src[e];
  int d = dst[e];
  float w = nrm[e];

  const float4 v = *(const float4*)(ein + (size_t)s * LATENT + c);
  float* o = eout + (size_t)d * LATENT + c;
  atomicAdd(o + 0, w * v.x);
  atomicAdd(o + 1, w * v.y);
  atomicAdd(o + 2, w * v.z);
  atomicAdd(o + 3, w * v.w);
}

// ---------------------------------------------------------------------------
// Final combine via V_WMMA_F32_16X16X4_F32, 4 chunks (64 elems) per wave.
// Per chunk:
//   B (4x16): row k = layer-k values for the 16 elements (the k-sum is
//             symmetric, so the (vgpr, lane-half) <-> k striping is irrelevant)
//   A (16x4): constant 0.25 in every slot (layout-independent)
//   D = A*B  -> every row = 0.25 * (e0+e1+e2+e3); row M=0 lives in c[0] of
//               lanes 0..15 with N = lane.
// The 4x unroll lets the scheduler hoist the B-loads of later chunks above
// earlier WMMAs (hides s_wait_loadcnt), and cuts grid size 4x.
// Also copies emb0 to the first half of d_out from the already-loaded B.x.
// ---------------------------------------------------------------------------
__global__ void combine_wmma_kernel(const float* __restrict__ e0, const float* __restrict__ e1,
                                    const float* __restrict__ e2, const float* __restrict__ e3,
                                    float* __restrict__ out_emb0, float* __restrict__ out_avg,
                                    int nchunks) {
  int nwaves = (int)((gridDim.x * blockDim.x) >> 5);
  int wave = (int)((blockIdx.x * blockDim.x + threadIdx.x) >> 5);
  int lane = (int)(threadIdx.x & 31u);
  int n = lane & 15;

  v2f a;
  a.x = 0.25f;
  a.y = 0.25f;

  for (int ch4 = wave * 4; ch4 < nchunks; ch4 += nwaves * 4) {
#pragma unroll
    for (int j = 0; j < 4; ++j) {
      int ch = ch4 + j;
      if (ch >= nchunks) break;  // wave-uniform (nchunks % 4 == 0 in practice)
      size_t base = (size_t)ch * 16;

      v2f b;
      if (lane < 16) {
        b.x = e0[base + n];
        b.y = e1[base + n];
      } else {
        b.x = e2[base + n];
        b.y = e3[base + n];
      }

      v8f c = {};
      // (neg_a, A, neg_b, B, c_mod, C, reuse_a, reuse_b)
      c = __builtin_amdgcn_wmma_f32_16x16x4_f32(false, a, false, b, (short)0, c, false, false);

      if (lane < 16) {
        out_avg[base + n] = c[0];   // D row M=0, N=lane
        out_emb0[base + n] = b.x;   // emb0 pass-through output
      }
    }
  }
}

// ---------------------------------------------------------------------------
// Launch
// ---------------------------------------------------------------------------
extern "C" void kernel_launch(void* const* d_in, const int* in_sizes, int n_in,
                              void* d_out, int out_size, void* d_ws, size_t ws_size,
                              hipStream_t stream) {
  const int* edge = (const int*)d_in[0];
  const float* emb0 = (const float*)d_in[1];
  const int E = in_sizes[0] / 2;
  const int N = in_sizes[1] / LATENT;
  const int NF = N * LATENT;

  const int* src = edge;
  const int* dst = edge + E;

  // Workspace layout (floats): deg[N] | e1[NF] | e2[NF] | e3[NF] | norm[E]
  float* ws = (float*)d_ws;
  float* deg = ws;
  float* e1 = deg + N;
  float* e2 = e1 + NF;
  float* e3 = e2 + NF;
  float* nrm = e3 + NF;

  float* out_emb0 = (float*)d_out;  // first output: emb0
  float* out_avg = out_emb0 + NF;   // second output: layer mean

  const int B = 256;  // 8 wave32s per block

  // Zero deg + e1 + e2 + e3 (contiguous region)
  int zcount = N + 3 * NF;
  zero_kernel<<<(zcount + B - 1) / B, B, 0, stream>>>(deg, zcount);

  degree_kernel<<<(E + B - 1) / B, B, 0, stream>>>(dst, deg, E);
  invsqrt_kernel<<<(N + B - 1) / B, B, 0, stream>>>(deg, N);
  norm_kernel<<<(E + B - 1) / B, B, 0, stream>>>(src, dst, deg, nrm, E);

  long long pt = (long long)E * 16;
  int pblocks = (int)((pt + B - 1) / B);
  propagate_kernel<<<pblocks, B, 0, stream>>>(emb0, e1, src, dst, nrm, E);
  propagate_kernel<<<pblocks, B, 0, stream>>>(e1, e2, src, dst, nrm, E);
  propagate_kernel<<<pblocks, B, 0, stream>>>(e2, e3, src, dst, nrm, E);

  int nchunks = NF / 16;           // LATENT=64 -> divisible; also % 4 == 0
  int chunks_per_wave = 4;
  int waves_needed = (nchunks + chunks_per_wave - 1) / chunks_per_wave;
  int wpb = B / 32;
  int cblocks = (waves_needed + wpb - 1) / wpb;
  combine_wmma_kernel<<<cblocks, B, 0, stream>>>(emb0, e1, e2, e3, out_emb0, out_avg, nchunks);
}